// MaterialModelFinder_7258494730315
// MI455X (gfx1250) — compile-verified
//
#include <hip/hip_runtime.h>

typedef __attribute__((ext_vector_type(2))) float v2f;
typedef __attribute__((ext_vector_type(4))) float v4f;
typedef __attribute__((ext_vector_type(8))) float v8f;

// Problem constants from the reference (n_time fixed at 128 by the harness).
#define TT     128
#define NGAUSS 8
#define NCOMP  6
#define EPB    16      // elements per block
#define NTHR   256     // 8 waves of 32

// out[e,g,t,c] = cumsum_t( p1 * disp[e,t] ), broadcast over g (8) and c (6).
// cumsum done as triangular GEMM with V_WMMA_F32_16X16X4_F32 (fp32-exact path).
__global__ __launch_bounds__(NTHR) void cumsum_bcast_wmma(
    const float* __restrict__ disp,
    const float* __restrict__ p1,
    float* __restrict__ out,
    int n_elem)
{
    __shared__ float s_disp[EPB * TT];   // 8 KB scaled input tile
    __shared__ float s_cum [EPB * TT];   // 8 KB cumsum tile

    const int tid   = threadIdx.x;
    const int ebase = blockIdx.x * EPB;
    const float scale = p1[0];

    // ---- Stage disp tile into LDS (coalesced b128 loads), scale by param_1 ----
    {
        const v4f* disp4 = reinterpret_cast<const v4f*>(disp) + (size_t)ebase * (TT / 4);
        v4f* sdisp4 = reinterpret_cast<v4f*>(s_disp);
        const int tile_f4 = EPB * TT / 4;   // 512
        #pragma unroll
        for (int q = tid; q < tile_f4; q += NTHR) {
            int e_local = q / (TT / 4);
            v4f v = {0.f, 0.f, 0.f, 0.f};
            if (ebase + e_local < n_elem) v = disp4[q];
            v *= scale;
            sdisp4[q] = v;
        }
        // Speculative prefetch of the next block's tile (global_prefetch_b8).
        if (tid < 32 && ebase + EPB < n_elem) {
            __builtin_prefetch(disp + (size_t)(ebase + EPB) * TT + tid * 64, 0, 0);
        }
    }
    __syncthreads();

    // ---- Wave 0: cum = disp_tile @ U  (U upper-triangular ones, 128x128) ----
    // Only K <= tbase+15 contributes to column tile at tbase (U zero above).
    if (tid < 32) {
        const int lane = tid;
        const int m  = lane & 15;   // A: row M;  B/C/D: column N
        const int hi = lane >> 4;   // K-half select for A/B, row+8 for C/D

        for (int j = 0; j < TT / 16; ++j) {
            const int tbase = j * 16;
            const int tcol  = tbase + m;           // this lane's output column
            v8f c = {};
            const int kend = tbase + 16;           // truncate at diagonal block
            for (int kbase = 0; kbase < kend; kbase += 4) {
                const int k0 = kbase + 2 * hi;
                // A fragment: 16x4 fp32, lane m holds K = k0, k0+1
                v2f a;
                a.x = s_disp[m * TT + k0];
                a.y = s_disp[m * TT + k0 + 1];
                // B fragment: 4x16 ones-below-diagonal, built from predicates
                v2f b;
                b.x = (k0     <= tcol) ? 1.0f : 0.0f;
                b.y = (k0 + 1 <= tcol) ? 1.0f : 0.0f;
                c = __builtin_amdgcn_wmma_f32_16x16x4_f32(
                        /*neg_a=*/false, a, /*neg_b=*/false, b,
                        /*c_mod=*/(short)0, c, /*reuse_a=*/false, /*reuse_b=*/false);
            }
            // C/D layout: VGPR v -> row (v + 8*hi), column (lane & 15)
            #pragma unroll
            for (int v = 0; v < 8; ++v) {
                s_cum[(v + 8 * hi) * TT + tbase + m] = c[v];
            }
        }
    }
    __syncthreads();

    // ---- Broadcast stores: each output value replicated 48x (8 g x 6 c) ----
    // Per (e,g) pair: 768 contiguous floats = 192 float4s. A float4 at offset
    // w = 4*w4 within the [t][c] run covers c-pattern {0..3}, {2..5} (one t)
    // or {4,5,0,1} (t and t+1)  ->  value pattern xxxx or xxyy.
    {
        const int T6_4   = TT * NCOMP / 4;              // 192
        const int total4 = EPB * NGAUSS * T6_4;         // 24576
        v4f* out4 = reinterpret_cast<v4f*>(out);
        const size_t base4 = (size_t)ebase * NGAUSS * T6_4;

        for (int Q = tid; Q < total4; Q += NTHR) {
            int pairIdx = Q / T6_4;                     // (e_local, g)
            int w4      = Q - pairIdx * T6_4;
            int e_local = pairIdx >> 3;                 // g is implicit (same data)
            if (ebase + e_local >= n_elem) break;       // Q monotonic per-thread
            int w  = 4 * w4;
            int t0 = w / 6;
            int r  = w - t0 * 6;                        // 0, 2, or 4
            float x = s_cum[e_local * TT + t0];
            v4f val;
            if (r == 4) {
                float y = s_cum[e_local * TT + t0 + 1];
                val = (v4f){x, x, y, y};
            } else {
                val = (v4f){x, x, x, x};
            }
            __builtin_nontemporal_store(val, &out4[base4 + Q]);
        }
    }
}

extern "C" void kernel_launch(void* const* d_in, const int* in_sizes, int n_in,
                              void* d_out, int out_size, void* d_ws, size_t ws_size,
                              hipStream_t stream) {
    (void)n_in; (void)out_size; (void)d_ws; (void)ws_size;
    const float* disp = (const float*)d_in[0];
    const float* p1   = (const float*)d_in[1];
    float* out        = (float*)d_out;

    const int n_time = in_sizes[2];            // 128 in the reference harness
    const int n_elem = in_sizes[0] / n_time;

    const int blocks = (n_elem + EPB - 1) / EPB;
    cumsum_bcast_wmma<<<blocks, NTHR, 0, stream>>>(disp, p1, out, n_elem);
}